// ParallelRY_51994874086071
// MI455X (gfx1250) — compile-verified
//
#include <hip/hip_runtime.h>

typedef float v2f __attribute__((ext_vector_type(2)));
typedef float v8f __attribute__((ext_vector_type(8)));

#define NQ        13
#define DIM       8192        // 2^13
#define NTHREADS  256         // 8 waves of 32 on gfx1250

// Gate element of a 4-qubit Kronecker block: G[r][n] = prod_j g_{q0+j}[r_{3-j}][n_{3-j}]
// with g = [[cos, -sin], [sin, cos]] (RY); qubit q0 is the MSB of the 4-bit group index.
__device__ __forceinline__ float gate_elem(const float* c, const float* s,
                                           int q0, int r, int n) {
    float v = 1.0f;
#pragma unroll
    for (int j = 0; j < 4; ++j) {
        const int rb = (r >> (3 - j)) & 1;
        const int nb = (n >> (3 - j)) & 1;
        const float cq = c[q0 + j];
        const float sq = s[q0 + j];
        v *= rb ? (nb ? cq : sq) : (nb ? -sq : cq);
    }
    return v;
}

// ---------------------------------------------------------------------------
// Run-once precompute (1 workgroup): builds the three 16x16 Kronecker gate
// blocks directly in per-lane WMMA fragment layout, plus cos/sin for qubit 0.
//   ws[g*256 + kc*64 + lane*2 + v] = G_g[4*kc + 2*(lane>>4) + v][lane&15]
//   ws[768] = cos(p0), ws[769] = sin(p0)
// All transcendental cost lives here, not in the bandwidth-bound kernel.
// ---------------------------------------------------------------------------
__global__ __launch_bounds__(NTHREADS)
void gate_precompute_kernel(const float* __restrict__ params,
                            float* __restrict__ ws)
{
    const int t = threadIdx.x;                 // 0..255 = kc*64 + lane*2 + v
    float c[NQ], s[NQ];
#pragma unroll
    for (int q = 0; q < NQ; ++q) {
        const float p = params[q];
        c[q] = cosf(p);
        s[q] = sinf(p);
    }

    const int v    = t & 1;
    const int lane = (t >> 1) & 31;
    const int kc   = t >> 6;
    const int r    = 4 * kc + 2 * (lane >> 4) + v;
    const int n    = lane & 15;

    ws[0 * 256 + t] = gate_elem(c, s, 9, r, n);   // qubits 9..12  (bits 3..0)
    ws[1 * 256 + t] = gate_elem(c, s, 5, r, n);   // qubits 5..8   (bits 7..4)
    ws[2 * 256 + t] = gate_elem(c, s, 1, r, n);   // qubits 1..4   (bits 11..8)
    if (t == 0) { ws[768] = c[0]; ws[769] = s[0]; }
}

// ---------------------------------------------------------------------------
// Main kernel: one batch row (32 KB) per workgroup, single HBM pass.
// Async global->LDS fill, 3 WMMA f32 stages + 1 butterfly, async LDS->global.
// ---------------------------------------------------------------------------
__global__ __launch_bounds__(NTHREADS)
void qkron_ry_kernel(const float* __restrict__ x,
                     const float* __restrict__ ws,
                     float* __restrict__ out)
{
    __shared__ float lds[DIM];                 // 32 KB

    const int tid  = threadIdx.x;
    const int lane = tid & 31;
    const int wave = tid >> 5;
    const int half = lane >> 4;
    const int ncol = lane & 15;

    // ---- async fill: 8 x B128 per thread, global -> LDS (no VGPR round trip)
    {
        const float* src = x + (size_t)blockIdx.x * DIM;
#pragma unroll
        for (int i = 0; i < DIM / 4 / NTHREADS; ++i) {
            const int idx = (tid + i * NTHREADS) * 4;
            const unsigned dsoff = (unsigned)(uintptr_t)(&lds[idx]);     // LDS aperture: addr[31:0]
            const unsigned long long ga = (unsigned long long)(uintptr_t)(src + idx);
            asm volatile("global_load_async_to_lds_b128 %0, %1, off"
                         :: "v"(dsoff), "v"(ga) : "memory");
        }
    }

    // ---- gate fragments: 3 coalesced b64 loads per lane per k-chunk (L2-hot)
    v2f ga4[4], gb4[4], gc4[4];
#pragma unroll
    for (int kc = 0; kc < 4; ++kc) {
        ga4[kc] = *(const v2f*)&ws[  0 + kc * 64 + lane * 2];
        gb4[kc] = *(const v2f*)&ws[256 + kc * 64 + lane * 2];
        gc4[kc] = *(const v2f*)&ws[512 + kc * 64 + lane * 2];
    }
    const float c0 = ws[768];
    const float s0 = ws[769];

    asm volatile("s_wait_asynccnt 0x0" ::: "memory");
    __syncthreads();

    // ------------- stage A: qubits 9..12 (column bits 3..0) -----------------
    // Contiguous 16x16 tiles: D = X_tile * G_A  (data = A operand).
#pragma unroll
    for (int tt = 0; tt < 4; ++tt) {
        const int base = (wave * 4 + tt) * 256;
        v8f acc = {};
#pragma unroll
        for (int kc = 0; kc < 4; ++kc) {
            const v2f a = *(const v2f*)&lds[base + ncol * 16 + 4 * kc + 2 * half];
            acc = __builtin_amdgcn_wmma_f32_16x16x4_f32(
                      false, a, false, ga4[kc], (short)0, acc, false, false);
        }
#pragma unroll
        for (int v = 0; v < 8; ++v)
            lds[base + (v + 8 * half) * 16 + ncol] = acc[v];
    }
    __syncthreads();

    // ------------- stage B: qubits 5..8 (column bits 7..4) ------------------
    // Row viewed as [h:32][m:16][l:16]: D = G_B^T * X_h (gate = A operand).
#pragma unroll
    for (int tt = 0; tt < 4; ++tt) {
        const int base = (wave * 4 + tt) * 256;
        v8f acc = {};
#pragma unroll
        for (int kc = 0; kc < 4; ++kc) {
            const int k0 = 4 * kc + 2 * half;
            v2f b;
            b.x = lds[base + (k0    ) * 16 + ncol];
            b.y = lds[base + (k0 + 1) * 16 + ncol];
            acc = __builtin_amdgcn_wmma_f32_16x16x4_f32(
                      false, gb4[kc], false, b, (short)0, acc, false, false);
        }
#pragma unroll
        for (int v = 0; v < 8; ++v)
            lds[base + (v + 8 * half) * 16 + ncol] = acc[v];
    }
    __syncthreads();

    // ------------- stage C: qubits 1..4 (column bits 11..8) -----------------
    // Row viewed as [h2:2][m:16][l:256]; l tiled in 16-wide chunks.
#pragma unroll
    for (int tt = 0; tt < 4; ++tt) {
        const int t    = wave * 4 + tt;
        const int base = (t >> 4) * 4096 + (t & 15) * 16;
        v8f acc = {};
#pragma unroll
        for (int kc = 0; kc < 4; ++kc) {
            const int k0 = 4 * kc + 2 * half;
            v2f b;
            b.x = lds[base + (k0    ) * 256 + ncol];
            b.y = lds[base + (k0 + 1) * 256 + ncol];
            acc = __builtin_amdgcn_wmma_f32_16x16x4_f32(
                      false, gc4[kc], false, b, (short)0, acc, false, false);
        }
#pragma unroll
        for (int v = 0; v < 8; ++v)
            lds[base + (v + 8 * half) * 256 + ncol] = acc[v];
    }
    __syncthreads();

    // ------------- qubit 0 (column bit 12): stride-4096 butterfly ------------
#pragma unroll
    for (int i = 0; i < (DIM / 2) / NTHREADS; ++i) {
        const int j  = tid + i * NTHREADS;
        const float x0 = lds[j];
        const float x1 = lds[j + DIM / 2];
        lds[j]           = c0 * x0 + s0 * x1;   // y0 =  cos*x0 + sin*x1
        lds[j + DIM / 2] = c0 * x1 - s0 * x0;   // y1 = -sin*x0 + cos*x1
    }
    __syncthreads();

    // ---- async drain: LDS -> global, 8 x B128 per thread --------------------
    {
        float* dst = out + (size_t)blockIdx.x * DIM;
#pragma unroll
        for (int i = 0; i < DIM / 4 / NTHREADS; ++i) {
            const int idx = (tid + i * NTHREADS) * 4;
            const unsigned dsoff = (unsigned)(uintptr_t)(&lds[idx]);
            const unsigned long long gaddr = (unsigned long long)(uintptr_t)(dst + idx);
            asm volatile("global_store_async_from_lds_b128 %0, %1, off"
                         :: "v"(gaddr), "v"(dsoff) : "memory");
        }
        asm volatile("s_wait_asynccnt 0x0" ::: "memory");
    }
}

extern "C" void kernel_launch(void* const* d_in, const int* in_sizes, int n_in,
                              void* d_out, int out_size, void* d_ws, size_t ws_size,
                              hipStream_t stream) {
    const float* x      = (const float*)d_in[0];   // (BATCH, 8192) f32
    const float* params = (const float*)d_in[1];   // (13,) f32
    float*       out    = (float*)d_out;           // (BATCH, 8192) f32
    float*       ws     = (float*)d_ws;            // needs 770 floats

    const int batch = in_sizes[0] / DIM;           // 512

    gate_precompute_kernel<<<1, NTHREADS, 0, stream>>>(params, ws);
    qkron_ry_kernel<<<batch, NTHREADS, 0, stream>>>(x, ws, out);

    (void)n_in; (void)out_size; (void)ws_size;
}